// CRDLoss_16509854286615
// MI455X (gfx1250) — compile-verified
//
#include <hip/hip_runtime.h>
#include <cstdint>
#include <cstddef>

// ---------------------------------------------------------------------------
// CRD-style contrastive distillation loss, forward only -> scalar.
// MI455X / gfx1250: f32 WMMA (V_WMMA_F32_16X16X4_F32) for all GEMMs,
// L2-resident (192MB) single copy of E = exp(G/tau) (134MB), one fused
// 19-class log pass per view.
// ---------------------------------------------------------------------------

typedef float v2f __attribute__((ext_vector_type(2)));
typedef float v8f __attribute__((ext_vector_type(8)));

#define HW_    4096
#define NB_    2
#define NCLS_  19
#define CF_    128
#define TAU_INV (1.0f / 0.07f)
#define PN_    (1.0f / (4096.0f * 2975.0f))
#define EPS_C  1e-7f
#define BN_EPS 1e-5f

// ---------------------------------------------------------------------------
// Generic 1x1-conv GEMM:  Y[b][o][hw] = sum_c W[o][c] * X[b][c][hw] + bias[o]
// One 16x16 output tile per wave; K stepped by 4 via f32 WMMA.
// f32 WMMA layout (ISA 7.12.2): A(16x4): lane L, vgpr j -> (M=L%16, K=2*(L/16)+j)
//                               B(4x16): lane L, vgpr j -> (K=2*(L/16)+j, N=L%16)
//                               C(16x16): lane L, vgpr v -> (M=v+8*(L/16), N=L%16)
// ---------------------------------------------------------------------------
__global__ void k_conv1x1_wmma(const float* __restrict__ W, const float* __restrict__ X,
                               const float* __restrict__ bias, float* __restrict__ Y,
                               int M, int C) {
  const int lane = threadIdx.x & 31;
  const int wv   = threadIdx.x >> 5;
  const int tiles_m = M >> 4;
  const int tiles   = tiles_m * ((NB_ * HW_) >> 4);
  const int t = blockIdx.x * (blockDim.x >> 5) + wv;
  if (t >= tiles) return;                     // wave-uniform exit (EXEC stays all-1)
  const int tm = t % tiles_m, tn = t / tiles_m;
  const int row0 = tm << 4;
  const int col0 = tn << 4;
  const int b = col0 / HW_, hw0 = col0 % HW_;
  const int ln = lane & 15, hi = lane >> 4;

  const float* Wp = W + (size_t)(row0 + ln) * C + (hi << 1);
  const float* Xp = X + (size_t)b * C * HW_ + (size_t)(hi << 1) * HW_ + hw0 + ln;

  v8f acc = {};
  for (int k = 0; k < C; k += 4) {
    v2f a, bb;
    a[0]  = Wp[k];
    a[1]  = Wp[k + 1];
    bb[0] = Xp[(size_t)k * HW_];
    bb[1] = Xp[(size_t)(k + 1) * HW_];
    acc = __builtin_amdgcn_wmma_f32_16x16x4_f32(false, a, false, bb, (short)0, acc,
                                                false, false);
  }
  float* Yp = Y + (size_t)b * M * HW_ + hw0 + ln;
#pragma unroll
  for (int v = 0; v < 8; ++v) {
    const int r = row0 + v + (hi << 3);
    Yp[(size_t)r * HW_] = acc[v] + bias[r];
  }
}

// ---------------------------------------------------------------------------
// Gram + exp epilogue:  E[b][p][q] = exp( (sum_c v1[b][c][p] * v2[b][c][q]) / tau )
// ---------------------------------------------------------------------------
__global__ void k_gram_exp_wmma(const float* __restrict__ V1, const float* __restrict__ V2,
                                float* __restrict__ E) {
  const int lane = threadIdx.x & 31;
  const int wv   = threadIdx.x >> 5;
  const int tiles_m = HW_ >> 4;               // 256
  const int tiles_b = tiles_m * tiles_m;
  const int t = blockIdx.x * (blockDim.x >> 5) + wv;
  if (t >= NB_ * tiles_b) return;
  const int b  = t / tiles_b;
  const int r  = t % tiles_b;
  const int tm = r % tiles_m, tn = r / tiles_m;
  const int p0 = tm << 4, q0 = tn << 4;
  const int ln = lane & 15, hi = lane >> 4;

  const float* Ap = V1 + (size_t)b * CF_ * HW_ + (size_t)(hi << 1) * HW_ + p0 + ln;
  const float* Bp = V2 + (size_t)b * CF_ * HW_ + (size_t)(hi << 1) * HW_ + q0 + ln;

  v8f acc = {};
#pragma unroll 4
  for (int k = 0; k < CF_; k += 4) {
    v2f a, bb;
    a[0]  = Ap[(size_t)k * HW_];
    a[1]  = Ap[(size_t)(k + 1) * HW_];
    bb[0] = Bp[(size_t)k * HW_];
    bb[1] = Bp[(size_t)(k + 1) * HW_];
    acc = __builtin_amdgcn_wmma_f32_16x16x4_f32(false, a, false, bb, (short)0, acc,
                                                false, false);
  }
  float* Ep = E + (size_t)b * HW_ * HW_ + q0 + ln;
#pragma unroll
  for (int v = 0; v < 8; ++v) {
    const int p = p0 + v + (hi << 3);
    Ep[(size_t)p * HW_] = __expf(acc[v] * TAU_INV);
  }
}

// ---------------------------------------------------------------------------
// BatchNorm training stats -> per-channel scale/shift (fold gamma/beta).
// ---------------------------------------------------------------------------
__global__ void k_bn_stats(const float* __restrict__ H, const float* __restrict__ gamma,
                           const float* __restrict__ beta, float* __restrict__ scale,
                           float* __restrict__ shift, int C) {
  __shared__ float s1[256], s2[256];
  const int c = blockIdx.x, tid = threadIdx.x;
  float sum = 0.f, sq = 0.f;
  for (int idx = tid; idx < NB_ * HW_; idx += 256) {
    const int b = idx >> 12, hw = idx & (HW_ - 1);
    const float v = H[((size_t)b * C + c) * HW_ + hw];
    sum += v; sq += v * v;
  }
  s1[tid] = sum; s2[tid] = sq; __syncthreads();
  for (int s = 128; s > 0; s >>= 1) {
    if (tid < s) { s1[tid] += s1[tid + s]; s2[tid] += s2[tid + s]; }
    __syncthreads();
  }
  if (tid == 0) {
    const float n   = (float)(NB_ * HW_);
    const float mu  = s1[0] / n;
    const float var = s2[0] / n - mu * mu;
    const float sc  = gamma[c] * rsqrtf(var + BN_EPS);
    scale[c] = sc;
    shift[c] = beta[c] - mu * sc;
  }
}

__global__ void k_bn_relu(float* __restrict__ H, const float* __restrict__ sc,
                          const float* __restrict__ sh, int C) {
  const size_t i = (size_t)blockIdx.x * blockDim.x + threadIdx.x;
  if (i >= (size_t)NB_ * C * HW_) return;
  const int c = (int)((i / HW_) % C);
  H[i] = fmaxf(0.f, H[i] * sc[c] + sh[c]);
}

// L2-normalize over the CF_ channels for each (b,hw) column, in place.
__global__ void k_l2norm(float* __restrict__ U) {
  const int j = blockIdx.x * blockDim.x + threadIdx.x;
  if (j >= NB_ * HW_) return;
  const int b = j >> 12, hw = j & (HW_ - 1);
  float* p = U + (size_t)b * CF_ * HW_ + hw;
  float s = 0.f;
  for (int c = 0; c < CF_; ++c) { const float v = p[(size_t)c * HW_]; s += v * v; }
  const float inv = 1.0f / fmaxf(sqrtf(s), 1e-12f);
  for (int c = 0; c < CF_; ++c) p[(size_t)c * HW_] *= inv;
}

// ---------------------------------------------------------------------------
// Row-wise (view12) sums of E and log(E):  r12[b,p] = sum_q E,  slog12 = sum_q log E
// ---------------------------------------------------------------------------
__global__ void k_rowsum_row(const float* __restrict__ E, float* __restrict__ rsum,
                             float* __restrict__ rlog) {
  __shared__ float s1[256], s2[256];
  const int bp = blockIdx.x;                  // b*HW + p
  const float* row = E + (size_t)bp * HW_;
  float se = 0.f, sl = 0.f;
  for (int q = threadIdx.x; q < HW_; q += 256) {
    const float e = row[q];
    se += e; sl += __logf(e);
  }
  s1[threadIdx.x] = se; s2[threadIdx.x] = sl; __syncthreads();
  for (int s = 128; s > 0; s >>= 1) {
    if (threadIdx.x < s) { s1[threadIdx.x] += s1[threadIdx.x + s]; s2[threadIdx.x] += s2[threadIdx.x + s]; }
    __syncthreads();
  }
  if (threadIdx.x == 0) { rsum[bp] = s1[0]; rlog[bp] = s2[0]; }
}

// Column-wise (view21, since E21 = E12^T): 16 columns per block, L2-resident reads.
__global__ void k_rowsum_col(const float* __restrict__ E, float* __restrict__ rsum,
                             float* __restrict__ rlog) {
  __shared__ float s1[256], s2[256];
  const int blk = blockIdx.x;
  const int b  = blk / (HW_ / 16);
  const int p0 = (blk % (HW_ / 16)) << 4;
  const int tx = threadIdx.x & 15, ty = threadIdx.x >> 4;
  const float* base = E + (size_t)b * HW_ * HW_ + p0 + tx;
  float se = 0.f, sl = 0.f;
  for (int k = 0; k < HW_ / 16; ++k) {
    const float e = base[(size_t)((k << 4) + ty) * HW_];
    se += e; sl += __logf(e);
  }
  s1[threadIdx.x] = se; s2[threadIdx.x] = sl; __syncthreads();
  if (ty == 0) {
    float a = 0.f, c = 0.f;
    for (int y = 0; y < 16; ++y) { a += s1[y * 16 + tx]; c += s2[y * 16 + tx]; }
    rsum[(size_t)b * HW_ + p0 + tx] = a;
    rlog[(size_t)b * HW_ + p0 + tx] = c;
  }
}

__global__ void k_tot(const float* __restrict__ r12, const float* __restrict__ r21,
                      float* __restrict__ tot12, float* __restrict__ tot21) {
  __shared__ float s1[256], s2[256];
  const int b = blockIdx.x;
  float a = 0.f, c = 0.f;
  for (int p = threadIdx.x; p < HW_; p += 256) { a += r12[b * HW_ + p]; c += r21[b * HW_ + p]; }
  s1[threadIdx.x] = a; s2[threadIdx.x] = c; __syncthreads();
  for (int s = 128; s > 0; s >>= 1) {
    if (threadIdx.x < s) { s1[threadIdx.x] += s1[threadIdx.x + s]; s2[threadIdx.x] += s2[threadIdx.x + s]; }
    __syncthreads();
  }
  if (threadIdx.x == 0) { tot12[b] = s1[0]; tot21[b] = s2[0]; }
}

// Nearest-neighbor downsample gt [N,1,512,512] -> gts [N,4096] (scale = 8 exactly).
__global__ void k_gts(const int* __restrict__ gt, int* __restrict__ gts) {
  const int i = blockIdx.x * blockDim.x + threadIdx.x;
  if (i >= NB_ * HW_) return;
  const int b = i >> 12, p = i & (HW_ - 1);
  const int h = p >> 6, w = p & 63;
  gts[i] = gt[(size_t)b * 512 * 512 + (size_t)(h << 3) * 512 + (w << 3)];
}

__global__ void k_init_counts(int* __restrict__ counts) {
  if (threadIdx.x < NCLS_) counts[threadIdx.x] = 0;
}

__global__ void k_hist(const int* __restrict__ gt, int* __restrict__ counts) {
  const int i = blockIdx.x * blockDim.x + threadIdx.x;
  if (i >= NB_ * 512 * 512) return;
  atomicAdd(&counts[gt[i]], 1);
}

// Per (class,b): negatives m, positive index (argmax of masked pos_rand, first-max
// tie-break like jnp.argmax), masked row-sum of r12/r21, then Z and K = c*Z.
__global__ void k_class_stats(const int* __restrict__ gts, const float* __restrict__ pos_rand,
                              const float* __restrict__ r12, const float* __restrict__ r21,
                              const float* __restrict__ tot12, const float* __restrict__ tot21,
                              int* __restrict__ posA, float* __restrict__ mA,
                              float* __restrict__ K12A, float* __restrict__ K21A,
                              float* __restrict__ Z12A, float* __restrict__ Z21A,
                              float* __restrict__ hasA) {
  __shared__ int   scnt[256];
  __shared__ float sbv[256];
  __shared__ int   sbi[256];
  __shared__ float sm12[256], sm21[256];
  const int i = blockIdx.x % NCLS_, b = blockIdx.x / NCLS_;
  const int tid = threadIdx.x;
  int cnt = 0, bi = 0x7fffffff;
  float bv = -1.0f, m12 = 0.f, m21 = 0.f;
  for (int p = tid; p < HW_; p += 256) {
    if (gts[b * HW_ + p] == i) {
      cnt++;
      const float pr = pos_rand[((size_t)i * NB_ + b) * HW_ + p];
      if (pr > bv || (pr == bv && p < bi)) { bv = pr; bi = p; }
      m12 += r12[b * HW_ + p];
      m21 += r21[b * HW_ + p];
    }
  }
  scnt[tid] = cnt; sbv[tid] = bv; sbi[tid] = bi; sm12[tid] = m12; sm21[tid] = m21;
  __syncthreads();
  for (int s = 128; s > 0; s >>= 1) {
    if (tid < s) {
      scnt[tid] += scnt[tid + s];
      sm12[tid] += sm12[tid + s];
      sm21[tid] += sm21[tid + s];
      if (sbv[tid + s] > sbv[tid] || (sbv[tid + s] == sbv[tid] && sbi[tid + s] < sbi[tid])) {
        sbv[tid] = sbv[tid + s]; sbi[tid] = sbi[tid + s];
      }
    }
    __syncthreads();
  }
  if (tid == 0) {
    const int   mc  = scnt[0];
    const float m   = (float)(HW_ - mc);
    const int   pos = (mc > 0) ? sbi[0] : 0;
    const float c   = m * PN_ + EPS_C;
    const float Z12 = (r12[b * HW_ + pos] + (tot12[b] - sm12[0])) / (m + 1.0f);
    const float Z21 = (r21[b * HW_ + pos] + (tot21[b] - sm21[0])) / (m + 1.0f);
    const int o = i * NB_ + b;
    posA[o] = pos; mA[o] = m;
    Z12A[o] = Z12; Z21A[o] = Z21;
    K12A[o] = c * Z12; K21A[o] = c * Z21;
    hasA[o] = (mc > 0) ? 1.f : 0.f;
  }
}

// ---------------------------------------------------------------------------
// The heavy kernels: fused 19-class row log-sums.
//   rowlog12[i][b][p] = sum_q log(E[b][p][q] + K12[i][b])      (view E12)
//   rowlog21[i][b][p] = sum_q log(E[b][q][p] + K21[i][b])      (view E21 = E^T)
// One pass over E per view; E (134MB) is L2-resident on the 192MB L2.
// ---------------------------------------------------------------------------
__global__ void k_rowlog_row(const float* __restrict__ E, const float* __restrict__ KA,
                             float* __restrict__ rowlog) {
  __shared__ float red[256];
  const int bp = blockIdx.x;                  // b*HW + p
  const int b = bp >> 12;
  float Kv[NCLS_];
#pragma unroll
  for (int i = 0; i < NCLS_; ++i) Kv[i] = KA[i * NB_ + b];
  float acc[NCLS_] = {};
  const float* row = E + (size_t)bp * HW_;
  for (int q = threadIdx.x; q < HW_; q += 256) {
    const float e = row[q];
#pragma unroll
    for (int i = 0; i < NCLS_; ++i) acc[i] += __logf(e + Kv[i]);
  }
  for (int i = 0; i < NCLS_; ++i) {
    red[threadIdx.x] = acc[i]; __syncthreads();
    for (int s = 128; s > 0; s >>= 1) {
      if (threadIdx.x < s) red[threadIdx.x] += red[threadIdx.x + s];
      __syncthreads();
    }
    if (threadIdx.x == 0) rowlog[((size_t)i * NB_ + b) * HW_ + (bp & (HW_ - 1))] = red[0];
    __syncthreads();
  }
}

__global__ void k_rowlog_col(const float* __restrict__ E, const float* __restrict__ KA,
                             float* __restrict__ rowlog) {
  __shared__ float red[256];
  const int blk = blockIdx.x;
  const int b  = blk / (HW_ / 16);
  const int p0 = (blk % (HW_ / 16)) << 4;
  const int tx = threadIdx.x & 15, ty = threadIdx.x >> 4;
  float Kv[NCLS_];
#pragma unroll
  for (int i = 0; i < NCLS_; ++i) Kv[i] = KA[i * NB_ + b];
  float acc[NCLS_] = {};
  const float* base = E + (size_t)b * HW_ * HW_ + p0 + tx;
  for (int k = 0; k < HW_ / 16; ++k) {
    const float e = base[(size_t)((k << 4) + ty) * HW_];
#pragma unroll
    for (int i = 0; i < NCLS_; ++i) acc[i] += __logf(e + Kv[i]);
  }
  for (int i = 0; i < NCLS_; ++i) {
    red[threadIdx.x] = acc[i]; __syncthreads();
    if (ty == 0) {
      float s = 0.f;
      for (int y = 0; y < 16; ++y) s += red[y * 16 + tx];
      rowlog[((size_t)i * NB_ + b) * HW_ + p0 + tx] = s;
    }
    __syncthreads();
  }
}

// Per (class,b): combine to per-batch per-view losses.
//   t1 = (sumLogE[pos] - rowlog[pos]) / HW
//   t2 = m*(log(m*Pn) + log(Z)) - (sum_p negm*rowlog)/HW
//   per = -(t1+t2)
__global__ void k_class_finalize(const int* __restrict__ gts,
                                 const float* __restrict__ rowlog12,
                                 const float* __restrict__ rowlog21,
                                 const float* __restrict__ slog12,
                                 const float* __restrict__ slog21,
                                 const int* __restrict__ posA, const float* __restrict__ mA,
                                 const float* __restrict__ Z12A, const float* __restrict__ Z21A,
                                 float* __restrict__ perA) {
  __shared__ float s1[256], s2[256];
  const int i = blockIdx.x % NCLS_, b = blockIdx.x / NCLS_;
  const int o = i * NB_ + b;
  const float* rl12 = rowlog12 + (size_t)o * HW_;
  const float* rl21 = rowlog21 + (size_t)o * HW_;
  float S12 = 0.f, S21 = 0.f;
  for (int p = threadIdx.x; p < HW_; p += 256) {
    const float ng = (gts[b * HW_ + p] == i) ? 0.f : 1.f;
    S12 += ng * rl12[p];
    S21 += ng * rl21[p];
  }
  s1[threadIdx.x] = S12; s2[threadIdx.x] = S21; __syncthreads();
  for (int s = 128; s > 0; s >>= 1) {
    if (threadIdx.x < s) { s1[threadIdx.x] += s1[threadIdx.x + s]; s2[threadIdx.x] += s2[threadIdx.x + s]; }
    __syncthreads();
  }
  if (threadIdx.x == 0) {
    const float m = mA[o];
    const int pos = posA[o];
    const float invHW = 1.0f / (float)HW_;
    const float lmp = __logf(m * PN_);
    const float t1_12 = (slog12[b * HW_ + pos] - rl12[pos]) * invHW;
    const float t2_12 = m * (lmp + __logf(Z12A[o])) - s1[0] * invHW;
    const float t1_21 = (slog21[b * HW_ + pos] - rl21[pos]) * invHW;
    const float t2_21 = m * (lmp + __logf(Z21A[o])) - s2[0] * invHW;
    perA[o * 2 + 0] = -(t1_21 + t2_21);     // s_loss view (E21)
    perA[o * 2 + 1] = -(t1_12 + t2_12);     // t_loss view (E12)
  }
}

__global__ void k_final(const float* __restrict__ perA, const float* __restrict__ hasA,
                        const int* __restrict__ counts, float* __restrict__ out) {
  if (threadIdx.x != 0 || blockIdx.x != 0) return;
  int np = 0;
  for (int i = 0; i < NCLS_; ++i) if (counts[i] > 0) np++;
  float total = 0.f;
  for (int i = 0; i < NCLS_; ++i) {
    if (counts[i] <= 0) continue;
    float l = 0.f;
    for (int v = 0; v < 2; ++v) {
      float nv = 0.f, s = 0.f;
      for (int b = 0; b < NB_; ++b) {
        const float h = hasA[i * NB_ + b];
        nv += h;
        s += h * perA[(i * NB_ + b) * 2 + v];
      }
      l += (nv > 0.f) ? s / nv : 0.f;
    }
    total += l;
  }
  out[0] = total / (float)np - 1.0f;
}

// ---------------------------------------------------------------------------
extern "C" void kernel_launch(void* const* d_in, const int* in_sizes, int n_in,
                              void* d_out, int out_size, void* d_ws, size_t ws_size,
                              hipStream_t stream) {
  (void)in_sizes; (void)n_in; (void)out_size; (void)ws_size;
  const float* f_s    = (const float*)d_in[0];
  const float* f_t    = (const float*)d_in[1];
  const int*   gt     = (const int*)d_in[2];
  const float* prand  = (const float*)d_in[3];
  const float* sW1    = (const float*)d_in[4];
  const float* sb1    = (const float*)d_in[5];
  const float* sgamma = (const float*)d_in[6];
  const float* sbeta  = (const float*)d_in[7];
  const float* sW2    = (const float*)d_in[8];
  const float* sb2    = (const float*)d_in[9];
  const float* tW1    = (const float*)d_in[10];
  const float* tb1    = (const float*)d_in[11];
  const float* tgamma = (const float*)d_in[12];
  const float* tbeta  = (const float*)d_in[13];
  const float* tW2    = (const float*)d_in[14];
  const float* tb2    = (const float*)d_in[15];

  char* ws = (char*)d_ws;
  size_t off = 0;
  auto alloc = [&](size_t bytes) -> void* {
    void* p = (void*)(ws + off);
    off += (bytes + 255) & ~(size_t)255;
    return p;
  };

  float* E    = (float*)alloc((size_t)NB_ * HW_ * HW_ * 4);   // 134 MB
  float* h_t  = (float*)alloc((size_t)NB_ * 512 * HW_ * 4);   // 16 MB
  float* h_s  = (float*)alloc((size_t)NB_ * 256 * HW_ * 4);   // 8 MB
  float* v1   = (float*)alloc((size_t)NB_ * CF_ * HW_ * 4);   // 4 MB
  float* v2   = (float*)alloc((size_t)NB_ * CF_ * HW_ * 4);   // 4 MB
  float* rl12 = (float*)alloc((size_t)NCLS_ * NB_ * HW_ * 4); // rowlog12
  float* rl21 = (float*)alloc((size_t)NCLS_ * NB_ * HW_ * 4); // rowlog21
  float* r12  = (float*)alloc((size_t)NB_ * HW_ * 4);
  float* r21  = (float*)alloc((size_t)NB_ * HW_ * 4);
  float* sl12 = (float*)alloc((size_t)NB_ * HW_ * 4);
  float* sl21 = (float*)alloc((size_t)NB_ * HW_ * 4);
  int*   gts  = (int*)  alloc((size_t)NB_ * HW_ * 4);
  float* scS  = (float*)alloc(256 * 4);
  float* shS  = (float*)alloc(256 * 4);
  float* scT  = (float*)alloc(512 * 4);
  float* shT  = (float*)alloc(512 * 4);
  float* tot12 = (float*)alloc(NB_ * 4);
  float* tot21 = (float*)alloc(NB_ * 4);
  int*   counts = (int*)alloc(NCLS_ * 4);
  int*   posA   = (int*)alloc(NCLS_ * NB_ * 4);
  float* mA     = (float*)alloc(NCLS_ * NB_ * 4);
  float* K12A   = (float*)alloc(NCLS_ * NB_ * 4);
  float* K21A   = (float*)alloc(NCLS_ * NB_ * 4);
  float* Z12A   = (float*)alloc(NCLS_ * NB_ * 4);
  float* Z21A   = (float*)alloc(NCLS_ * NB_ * 4);
  float* hasA   = (float*)alloc(NCLS_ * NB_ * 4);
  float* perA   = (float*)alloc(NCLS_ * NB_ * 2 * 4);

  // ---- student embedding ----
  k_conv1x1_wmma<<<(16 * 512) / 8, 256, 0, stream>>>(sW1, f_s, sb1, h_s, 256, 256);
  k_bn_stats<<<256, 256, 0, stream>>>(h_s, sgamma, sbeta, scS, shS, 256);
  k_bn_relu<<<(NB_ * 256 * HW_) / 256, 256, 0, stream>>>(h_s, scS, shS, 256);
  k_conv1x1_wmma<<<(8 * 512) / 8, 256, 0, stream>>>(sW2, h_s, sb2, v1, 128, 256);
  k_l2norm<<<(NB_ * HW_) / 256, 256, 0, stream>>>(v1);

  // ---- teacher embedding ----
  k_conv1x1_wmma<<<(32 * 512) / 8, 256, 0, stream>>>(tW1, f_t, tb1, h_t, 512, 512);
  k_bn_stats<<<512, 256, 0, stream>>>(h_t, tgamma, tbeta, scT, shT, 512);
  k_bn_relu<<<(NB_ * 512 * HW_) / 256, 256, 0, stream>>>(h_t, scT, shT, 512);
  k_conv1x1_wmma<<<(8 * 512) / 8, 256, 0, stream>>>(tW2, h_t, tb2, v2, 128, 512);
  k_l2norm<<<(NB_ * HW_) / 256, 256, 0, stream>>>(v2);

  // ---- Gram + exp ----
  k_gram_exp_wmma<<<(NB_ * 256 * 256) / 8, 256, 0, stream>>>(v1, v2, E);

  // ---- row/col sums of E and logE ----
  k_rowsum_row<<<NB_ * HW_, 256, 0, stream>>>(E, r12, sl12);
  k_rowsum_col<<<NB_ * (HW_ / 16), 256, 0, stream>>>(E, r21, sl21);
  k_tot<<<NB_, 256, 0, stream>>>(r12, r21, tot12, tot21);

  // ---- labels, class presence, per-class stats ----
  k_gts<<<(NB_ * HW_) / 256, 256, 0, stream>>>(gt, gts);
  k_init_counts<<<1, 32, 0, stream>>>(counts);
  k_hist<<<(NB_ * 512 * 512) / 256, 256, 0, stream>>>(gt, counts);
  k_class_stats<<<NCLS_ * NB_, 256, 0, stream>>>(gts, prand, r12, r21, tot12, tot21,
                                                 posA, mA, K12A, K21A, Z12A, Z21A, hasA);

  // ---- heavy fused 19-class log passes (one per view) ----
  k_rowlog_row<<<NB_ * HW_, 256, 0, stream>>>(E, K12A, rl12);
  k_rowlog_col<<<NB_ * (HW_ / 16), 256, 0, stream>>>(E, K21A, rl21);

  // ---- combine ----
  k_class_finalize<<<NCLS_ * NB_, 256, 0, stream>>>(gts, rl12, rl21, sl12, sl21,
                                                    posA, mA, Z12A, Z21A, perA);
  k_final<<<1, 1, 0, stream>>>(perA, hasA, counts, (float*)d_out);
}